// AGFN_5128190951752
// MI455X (gfx1250) — compile-verified
//
#include <hip/hip_runtime.h>
#include <math.h>

typedef __attribute__((ext_vector_type(2))) float v2f;
typedef __attribute__((ext_vector_type(8))) float v8f;

#define NUc   100000
#define NBc   50000
#define NIc   100000
#define Dc    64
#define Lc    3
#define BATCHc 8192
#define N_UBc (NUc + NBc)   // 150000
#define N_IUc (NIc + NUc)   // 200000
#define LEAKYc 0.2f

// ---------------------------------------------------------------------------
// init: x = acc = concat(emb_u, emb_b), vectorized float4
// ---------------------------------------------------------------------------
__global__ void agfn_init_kernel(const float4* __restrict__ eu,
                                 const float4* __restrict__ eb,
                                 float4* __restrict__ x,
                                 float4* __restrict__ acc) {
    int i = blockIdx.x * blockDim.x + threadIdx.x;
    const int n4  = N_UBc * (Dc / 4);
    const int nu4 = NUc * (Dc / 4);
    if (i >= n4) return;
    float4 v = (i < nu4) ? eu[i] : eb[i - nu4];
    x[i]   = v;
    acc[i] = v;
}

// ---------------------------------------------------------------------------
// SpMM scatter: out[dst[e]] += vals[e] * (filter? leaky(fw[src[e]]):1) * x[src[e]]
// 16 lanes per edge, each lane owns one float4 slice of the 64-float row.
// Coalesced 256B gather, 4 hardware f32 atomics per lane to L2.
// ---------------------------------------------------------------------------
__global__ void agfn_spmm_kernel(const float* __restrict__ vals,
                                 const int*   __restrict__ src,
                                 const int*   __restrict__ dst,
                                 const float* __restrict__ xin,
                                 const float* __restrict__ fw,   // filter row or nullptr
                                 float*       __restrict__ yout,
                                 int nnz) {
    long long tid = (long long)blockIdx.x * blockDim.x + threadIdx.x;
    int e  = (int)(tid >> 4);
    int c4 = (int)(tid & 15);
    if (e >= nnz) return;

    int   s = src[e];
    int   d = dst[e];
    float scale = vals[e];
    if (fw != nullptr) {
        float w = fw[s];
        scale *= (w < 0.0f) ? (LEAKYc * w) : w;
    }
    float4 xv = ((const float4*)(xin + (size_t)s * Dc))[c4];
    float* yp = yout + (size_t)d * Dc + (size_t)c4 * 4;
    unsafeAtomicAdd(yp + 0, scale * xv.x);
    unsafeAtomicAdd(yp + 1, scale * xv.y);
    unsafeAtomicAdd(yp + 2, scale * xv.z);
    unsafeAtomicAdd(yp + 3, scale * xv.w);
}

// ---------------------------------------------------------------------------
// acc += x_new  (float4)
// ---------------------------------------------------------------------------
__global__ void agfn_accadd_kernel(float4* __restrict__ acc,
                                   const float4* __restrict__ xn,
                                   int n4) {
    int i = blockIdx.x * blockDim.x + threadIdx.x;
    if (i >= n4) return;
    float4 a = acc[i], b = xn[i];
    a.x += b.x; a.y += b.y; a.z += b.z; a.w += b.w;
    acc[i] = a;
}

// ---------------------------------------------------------------------------
// Rating + softplus loss via V_WMMA_F32_16X16X4_F32.
// One wave handles 16 batch items: A = 16 user rows (16xK), B = 16 bundle
// rows as KxN, accumulate 16 k-steps of K=4 over D=64 into a 16x16 f32 tile;
// the diagonal holds the per-item dot products.
//
// A 16x4 f32 layout: lanes 0-15 hold M=lane {K0 in v0, K1 in v1},
//                    lanes 16-31 hold M=lane-16 {K2, K3}.
// B 4x16 layout mirrors it (rows striped across lanes per VGPR), so both
// operands are a float2 load at d = 4*kk + 2*(lane>>4) from row (lane&15).
// C/D diag: m<8 -> VGPR m @ lane m ; m>=8 -> VGPR m-8 @ lane m+16.
// ---------------------------------------------------------------------------
__global__ void agfn_loss_wmma_kernel(const float* __restrict__ acc,
                                      const int*   __restrict__ u_idx,
                                      const int*   __restrict__ b_idx,
                                      float*       __restrict__ out) {
    int lane = threadIdx.x & 31;
    int wave = (blockIdx.x * blockDim.x + threadIdx.x) >> 5;
    int base = wave * 16;                 // grid sized so base < BATCH always
    int r    = base + (lane & 15);
    int hi   = lane >> 4;                 // 0 or 1

    const float* urow = acc + (size_t)u_idx[r] * Dc;
    const float* b0r  = acc + (size_t)(NUc + b_idx[2 * r + 0]) * Dc;
    const float* b1r  = acc + (size_t)(NUc + b_idx[2 * r + 1]) * Dc;

    v8f c0 = {}; v8f c1 = {};
#pragma unroll
    for (int kk = 0; kk < 16; ++kk) {
        int dd = kk * 4 + hi * 2;
        v2f a;  a.x  = urow[dd]; a.y  = urow[dd + 1];
        v2f b0; b0.x = b0r[dd];  b0.y = b0r[dd + 1];
        v2f b1; b1.x = b1r[dd];  b1.y = b1r[dd + 1];
        c0 = __builtin_amdgcn_wmma_f32_16x16x4_f32(false, a, false, b0,
                                                   (short)0, c0, false, false);
        c1 = __builtin_amdgcn_wmma_f32_16x16x4_f32(false, a, false, b1,
                                                   (short)0, c1, false, false);
    }

    float d0[8], d1[8];
#pragma unroll
    for (int i = 0; i < 8; ++i) { d0[i] = c0[i]; d1[i] = c1[i]; }

    float r0 = 0.0f, r1 = 0.0f;
    int   have = 0;
    if (lane < 8)        { r0 = d0[lane];      r1 = d1[lane];      have = 1; }
    else if (lane >= 24) { r0 = d0[lane - 24]; r1 = d1[lane - 24]; have = 1; }

    if (have) {
        // final = acc/4 on both sides -> rating diff = (dot1 - dot0)/16
        float diff = (r1 - r0) * (1.0f / 16.0f);
        float sp = (diff > 0.0f) ? diff + log1pf(expf(-diff))
                                 : log1pf(expf(diff));
        unsafeAtomicAdd(out, sp * (1.0f / (float)BATCHc));
    }
}

// ---------------------------------------------------------------------------
// l2 = 0.5*(sum(emb_u^2)+sum(emb_b^2))/NU  -> out[1]
// ---------------------------------------------------------------------------
__global__ void agfn_l2_kernel(const float* __restrict__ eu,
                               const float* __restrict__ eb,
                               float* __restrict__ out) {
    const size_t n1 = (size_t)NUc * Dc;
    const size_t n2 = (size_t)NBc * Dc;
    size_t stride = (size_t)gridDim.x * blockDim.x;
    float s = 0.0f;
    for (size_t i = (size_t)blockIdx.x * blockDim.x + threadIdx.x;
         i < n1 + n2; i += stride) {
        float v = (i < n1) ? eu[i] : eb[i - n1];
        s += v * v;
    }
#pragma unroll
    for (int o = 16; o > 0; o >>= 1) s += __shfl_down(s, o, 32);
    if ((threadIdx.x & 31) == 0)
        unsafeAtomicAdd(out + 1, s * (0.5f / (float)NUc));
}

// ---------------------------------------------------------------------------
// Host orchestration
// ---------------------------------------------------------------------------
extern "C" void kernel_launch(void* const* d_in, const int* in_sizes, int n_in,
                              void* d_out, int out_size, void* d_ws, size_t ws_size,
                              hipStream_t stream) {
    const float* emb_u    = (const float*)d_in[0];
    const float* emb_b    = (const float*)d_in[1];
    const float* filter_w = (const float*)d_in[2];
    const float* vals     = (const float*)d_in[3];
    const int*   rows     = (const int*)d_in[4];
    const int*   cols     = (const int*)d_in[5];
    const int*   u_idx    = (const int*)d_in[6];
    const int*   b_idx    = (const int*)d_in[7];
    float*       out      = (float*)d_out;

    const int nnz = in_sizes[3];

    const size_t XF = (size_t)N_UBc * Dc;   // 9.6M floats
    const size_t YF = (size_t)N_IUc * Dc;   // 12.8M floats
    float* acc = (float*)d_ws;
    float* x0  = acc + XF;
    float* x1  = x0 + XF;
    float* y   = x1 + XF;

    hipMemsetAsync(out, 0, 2 * sizeof(float), stream);

    // init x0 and acc
    {
        int n4 = N_UBc * (Dc / 4);
        int blocks = (n4 + 255) / 256;
        agfn_init_kernel<<<blocks, 256, 0, stream>>>(
            (const float4*)emb_u, (const float4*)emb_b, (float4*)x0, (float4*)acc);
    }

    float* xc = x0;
    float* xn = x1;
    long long spmm_threads = (long long)nnz * 16;
    int spmm_blocks = (int)((spmm_threads + 255) / 256);
    int n4 = N_UBc * (Dc / 4);
    int add_blocks = (n4 + 255) / 256;

    for (int l = 0; l < Lc; ++l) {
        hipMemsetAsync(y, 0, YF * sizeof(float), stream);
        // y[col] += v * x[row]
        agfn_spmm_kernel<<<spmm_blocks, 256, 0, stream>>>(
            vals, rows, cols, xc, nullptr, y, nnz);
        hipMemsetAsync(xn, 0, XF * sizeof(float), stream);
        // x'[row] += v * leaky(filter[l][col]) * y[col]   (filter fused at src)
        agfn_spmm_kernel<<<spmm_blocks, 256, 0, stream>>>(
            vals, cols, rows, y, filter_w + (size_t)l * N_IUc, xn, nnz);
        agfn_accadd_kernel<<<add_blocks, 256, 0, stream>>>(
            (float4*)acc, (const float4*)xn, n4);
        float* t = xc; xc = xn; xn = t;
    }

    // rating + softplus loss: 16 items per wave -> BATCH/16 = 512 waves
    {
        int waves  = BATCHc / 16;          // 512
        int threads = waves * 32;          // 16384
        int blocks  = threads / 128;       // 128 blocks of 4 waves
        agfn_loss_wmma_kernel<<<blocks, 128, 0, stream>>>(acc, u_idx, b_idx, out);
    }

    // l2 regularizer
    agfn_l2_kernel<<<512, 256, 0, stream>>>(emb_u, emb_b, out);
}